// DynamicalModel_28527172780723
// MI455X (gfx1250) — compile-verified
//
#include <hip/hip_runtime.h>

// ---------------------------------------------------------------------------
// DynamicalModel on MI455X (gfx1250, wave32, WMMA)
//
// Strategy:
//   out[b,t,k,:] is nonzero only for k = argmax(u_t[b]).  So:
//     1) zero d_out with nontemporal 128-bit stores (256 MB stream, ~11us floor)
//     2) compute labels + compact batches per label (one tiny block)
//     3) expH[k] = Taylor-10 exp of the Hamiltonian, f32 WMMA GEMMs
//     4) propagate: per (k, group-of-16-batches), 63 steps of
//        S_{t+1} = expH[k] @ S_t as 128x16x128 f32 WMMA GEMMs, A-fragments
//        resident in VGPRs, state ping-pong in LDS, coalesced output scatter.
// All f32 (matches reference precision; 63-step chain would drift in f16).
// ---------------------------------------------------------------------------

typedef __attribute__((ext_vector_type(2))) float v2f;
typedef __attribute__((ext_vector_type(4))) float v4f;
typedef __attribute__((ext_vector_type(8))) float v8f;

#define D2     128          // 2*SD
#define NK     16           // operators
#define NB     512          // batch
#define NT     64           // timesteps
#define MAT    (D2 * D2)    // 16384 floats per matrix
#define SPITCH 20           // LDS row pitch (floats) for state blocks

// V_WMMA_F32_16X16X4_F32 : D(16x16 f32) = A(16x4 f32) x B(4x16 f32) + C
// 8-arg signature pattern (probe-confirmed arity for 16x16x4 group).
__device__ __forceinline__ v8f wmma4(v2f a, v2f b, v8f c) {
  return __builtin_amdgcn_wmma_f32_16x16x4_f32(
      /*neg_a=*/false, a, /*neg_b=*/false, b,
      /*c_mod=*/(short)0, c, /*reuse_a=*/false, /*reuse_b=*/false);
}

// A-fragment (16x4, f32): lanes 0-15 -> {K=0,K=1}, lanes 16-31 -> {K=2,K=3},
// row M = lane%16.  8-byte aligned -> single b64 load.
__device__ __forceinline__ v2f load_afrag(const float* M, int rowBase, int kc,
                                          int lane) {
  int row = rowBase + (lane & 15);
  int col = kc * 4 + ((lane >> 4) << 1);
  return *(const v2f*)(M + row * D2 + col);
}

// ---------------------------------------------------------------------------
// 1) zero output (nontemporal: bulk of it is never touched again)
// ---------------------------------------------------------------------------
__global__ void zero_kernel(v4f* __restrict__ out, int n4) {
  int i = blockIdx.x * blockDim.x + threadIdx.x;
  int stride = gridDim.x * blockDim.x;
  v4f z = {0.f, 0.f, 0.f, 0.f};
  for (; i < n4; i += stride) __builtin_nontemporal_store(z, &out[i]);
}

// ---------------------------------------------------------------------------
// 2) labels = argmax(u_t, axis=1); compact batch ids per label
// ---------------------------------------------------------------------------
__global__ __launch_bounds__(NB) void labels_kernel(
    const float* __restrict__ u_t, int* __restrict__ counts,
    int* __restrict__ perm) {
  __shared__ int lc[NK];
  int b = threadIdx.x;
  if (b < NK) lc[b] = 0;
  __syncthreads();
  const float* u = u_t + b * NK;
  int best = 0;
  float bv = u[0];
#pragma unroll
  for (int k = 1; k < NK; ++k) {
    float v = u[k];
    if (v > bv) { bv = v; best = k; }   // strict '>' keeps first max (argmax)
  }
  int slot = atomicAdd(&lc[best], 1);
  perm[best * NB + slot] = b;
  __syncthreads();
  if (b < NK) counts[b] = lc[b];
}

// ---------------------------------------------------------------------------
// 3) expH[k] = I + sum_{i=1..10} X^i / i!,  X = S @ sym(H[k])
//    One block per k, 8 waves; wave r owns row-tile r (8 col-tiles, acc in
//    VGPRs).  Power matrix ping-pongs through global scratch (L2-hot).
// ---------------------------------------------------------------------------
__global__ __launch_bounds__(256) void expm_kernel(
    const float* __restrict__ H, float* __restrict__ GHm,
    float* __restrict__ Pa, float* __restrict__ Pb, float* __restrict__ GE) {
  int k = blockIdx.x;
  const float* Hk = H + k * MAT;
  float* Hm = GHm + k * MAT;
  float* P0 = Pa + k * MAT;
  float* P1 = Pb + k * MAT;
  float* E  = GE + k * MAT;

  int tid = threadIdx.x;
  // Build X = S @ sym(H): rows 0..63 = A[64+i,:], rows 64..127 = -A[i-64,:]
  for (int idx = tid; idx < MAT; idx += 256) {
    int i = idx >> 7, j = idx & 127;
    int src = (i < 64) ? (i + 64) : (i - 64);
    float a = 0.5f * (Hk[src * D2 + j] + Hk[j * D2 + src]);
    Hm[idx] = (i < 64) ? a : -a;
  }
  __syncthreads();  // block-scope visibility of Hm

  int wave = tid >> 5;
  int lane = tid & 31;
  int lhalf = lane >> 4;       // 0 | 1  (C/D rows +0 | +8)
  int l15  = lane & 15;
  int rt   = wave;             // row tile 0..7

  // acc = I + X  (covers the i=1 Taylor term)
  v8f acc[8];
#pragma unroll
  for (int ct = 0; ct < 8; ++ct) {
#pragma unroll
    for (int v = 0; v < 8; ++v) {
      int row = rt * 16 + v + lhalf * 8;
      int col = ct * 16 + l15;
      acc[ct][v] = Hm[row * D2 + col] + ((row == col) ? 1.0f : 0.0f);
    }
  }

  const float* Pcur = Hm;   // X^1/1!
  float* Pnext = P0;
  for (int it = 2; it <= 10; ++it) {
    float inv = 1.0f / (float)it;
    v8f d[8];
#pragma unroll
    for (int ct = 0; ct < 8; ++ct) d[ct] = (v8f){0,0,0,0,0,0,0,0};

    for (int kc = 0; kc < 32; ++kc) {
      v2f a = load_afrag(Pcur, rt * 16, kc, lane);
#pragma unroll
      for (int ct = 0; ct < 8; ++ct) {
        // B-fragment (4x16): lanes 0-15 rows K=0/1, lanes 16-31 rows K=2/3
        int brow = kc * 4 + lhalf * 2;
        int bcol = ct * 16 + l15;
        v2f b;
        b.x = Pnext == Pnext ? Hm[brow * D2 + bcol] : 0.f;  // Hm[brow][bcol]
        b.y = Hm[(brow + 1) * D2 + bcol];
        d[ct] = wmma4(a, b, d[ct]);
      }
    }
    // newP = d*inv ; acc += newP ; store newP
#pragma unroll
    for (int ct = 0; ct < 8; ++ct) {
#pragma unroll
      for (int v = 0; v < 8; ++v) {
        float nv = d[ct][v] * inv;
        acc[ct][v] += nv;
        int row = rt * 16 + v + lhalf * 8;
        int col = ct * 16 + l15;
        Pnext[row * D2 + col] = nv;
      }
    }
    __syncthreads();  // all stores done before anyone reads Pnext next iter
    const float* t = Pnext;
    Pnext = (Pnext == P0) ? P1 : P0;
    Pcur = t;
  }
  // E = acc
#pragma unroll
  for (int ct = 0; ct < 8; ++ct) {
#pragma unroll
    for (int v = 0; v < 8; ++v) {
      int row = rt * 16 + v + lhalf * 8;
      int col = ct * 16 + l15;
      E[row * D2 + col] = acc[ct][v];
    }
  }
}

// ---------------------------------------------------------------------------
// 4) propagation: block = (k, group of 16 batches with label k).
//    wave r holds rows [16r,16r+16) of expH[k] as 32 v2f A-fragments in
//    VGPRs for the whole 63-step loop; state ping-pongs in LDS.
// ---------------------------------------------------------------------------
__global__ __launch_bounds__(256) void prop_kernel(
    const float* __restrict__ s_t, const float* __restrict__ GE,
    const int* __restrict__ perm, const int* __restrict__ counts,
    float* __restrict__ out) {
  int k = blockIdx.x;
  int g = blockIdx.y;
  int nk = counts[k];
  if (g * 16 >= nk) return;  // uniform per block

  __shared__ float S0[D2 * SPITCH];
  __shared__ float S1[D2 * SPITCH];
  __shared__ int bcol[16];

  int tid = threadIdx.x;
  int wave = tid >> 5;       // 0..7 : row tile
  int lane = tid & 31;
  int lhalf = lane >> 4;
  int l15 = lane & 15;

  if (tid < 16) {
    int idx = g * 16 + tid;
    bcol[tid] = (idx < nk) ? perm[k * NB + idx] : -1;
  }
  __syncthreads();

  // resident A-fragments: rows [wave*16, wave*16+16) of expH[k], all K
  const float* M = GE + k * MAT;
  v2f A[32];
#pragma unroll
  for (int kc = 0; kc < 32; ++kc) A[kc] = load_afrag(M, wave * 16, kc, lane);

  // load initial state block (columns = batches), pitch SPITCH vs banks
  for (int idx = tid; idx < 16 * D2; idx += 256) {
    int col = idx >> 7, row = idx & 127;
    int b = bcol[col];
    S0[row * SPITCH + col] = (b >= 0) ? s_t[(b * NK + k) * D2 + row] : 0.f;
  }
  __syncthreads();

  float* Scur = S0;
  float* Snext = S1;

  // t = 0 : emit initial state
  for (int idx = tid; idx < 16 * D2; idx += 256) {
    int col = idx >> 7, row = idx & 127;
    int b = bcol[col];
    if (b >= 0) out[((b * NT + 0) * NK + k) * D2 + row] = Scur[row * SPITCH + col];
  }

  for (int t = 1; t < NT; ++t) {
    v8f acc = (v8f){0,0,0,0,0,0,0,0};
#pragma unroll
    for (int kc = 0; kc < 32; ++kc) {
      int brow = kc * 4 + lhalf * 2;
      v2f b;
      b.x = Scur[brow * SPITCH + l15];
      b.y = Scur[(brow + 1) * SPITCH + l15];
      acc = wmma4(A[kc], b, acc);
    }
    // D layout: VGPR v -> row wave*16 + v + (lane>=16 ? 8 : 0), col = lane%16
    {
      int rbase = wave * 16 + lhalf * 8;
#pragma unroll
      for (int v = 0; v < 8; ++v) Snext[(rbase + v) * SPITCH + l15] = acc[v];
    }
    __syncthreads();  // Snext complete; old Scur free for next iter's stores

    for (int idx = tid; idx < 16 * D2; idx += 256) {
      int col = idx >> 7, row = idx & 127;
      int b = bcol[col];
      if (b >= 0)
        out[((b * NT + t) * NK + k) * D2 + row] = Snext[row * SPITCH + col];
    }
    float* tmp = Scur; Scur = Snext; Snext = tmp;
  }
}

// ---------------------------------------------------------------------------
extern "C" void kernel_launch(void* const* d_in, const int* in_sizes, int n_in,
                              void* d_out, int out_size, void* d_ws,
                              size_t ws_size, hipStream_t stream) {
  const float* u_t = (const float*)d_in[0];   // [512,16]
  const float* s_t = (const float*)d_in[1];   // [512,16,128]
  const float* H   = (const float*)d_in[2];   // [16,128,128]
  float* out = (float*)d_out;                 // [512,64,16,128]

  // workspace carve (~4.23 MB)
  float* GHm = (float*)d_ws;
  float* Pa  = GHm + NK * MAT;
  float* Pb  = Pa  + NK * MAT;
  float* GE  = Pb  + NK * MAT;
  int* counts = (int*)(GE + NK * MAT);
  int* perm   = counts + NK;

  int n4 = out_size / 4;  // 16,777,216 float4 stores
  zero_kernel<<<(n4 + 255) / 256, 256, 0, stream>>>((v4f*)out, n4);
  labels_kernel<<<1, NB, 0, stream>>>(u_t, counts, perm);
  expm_kernel<<<NK, 256, 0, stream>>>(H, GHm, Pa, Pb, GE);
  prop_kernel<<<dim3(NK, NB / 16), 256, 0, stream>>>(s_t, GE, perm, counts, out);
}